// HugeNet_65850438582499
// MI455X (gfx1250) — compile-verified
//
#include <hip/hip_runtime.h>
#include <stdint.h>

typedef __attribute__((ext_vector_type(2))) float v2f;
typedef __attribute__((ext_vector_type(8))) float v8f;

#define NL        10000
#define D         100
#define DD        (D * D)
#define DOUT      10
#define MTILE     16
#define DP        112     // N padded to 7*16 (output-column tiles)
#define KSTEP     25      // D / 4 exactly -- no K padding needed
#define WS        113     // LDS row stride for transposed W (odd -> conflict-free)
#define WROWS     100     // only k < 100 rows of W^T are ever read
#define HS        116     // LDS row stride for activations (mult of 4 -> aligned b64 A-frags)
#define NTHREADS  224     // 7 wave32 -> one 16x16 N-tile per wave
#define STAGE_ITERS ((DD + NTHREADS - 1) / NTHREADS)   // 45 async b32 per thread per layer

#define WBUF_FLOATS (WROWS * WS)
#define HBUF_FLOATS (MTILE * HS)
#define HB_BASE     (2 * WBUF_FLOATS)
#define SMEM_FLOATS (2 * WBUF_FLOATS + 2 * HBUF_FLOATS)

static_assert(STAGE_ITERS * NTHREADS >= DD, "staging coverage");

// Issue one layer's weights as async DMA: global W[j][k] -> LDS wt[k*WS + j]
// (transposed scatter). Tracked by ASYNCcnt; runs under the WMMA phase.
__device__ __forceinline__ void stage_async(uint64_t gbase, uint32_t lds_base_b, int t) {
  int e = t;            // element index within the 100x100 layer
  int j = t / D;        // output row of W  (column of wt)
  int k = t - j * D;    // input col of W   (row of wt)
  for (int i = 0; i < STAGE_ITERS; ++i) {
    if (e < DD) {
      uint32_t lb = lds_base_b + (uint32_t)(k * WS + j) * 4u;
      uint32_t go = (uint32_t)e * 4u;
      asm volatile("global_load_async_to_lds_b32 %0, %1, %2"
                   :: "v"(lb), "v"(go), "s"(gbase) : "memory");
    }
    // e += 224 = 2*100 + 24: advance (j,k) branchlessly
    e += NTHREADS;
    k += 24; j += 2;
    if (k >= D) { k -= D; j += 1; }
  }
}

__global__ __launch_bounds__(NTHREADS) void hugenet_kernel(
    const float* __restrict__ x, const float* __restrict__ W,
    const float* __restrict__ b, const float* __restrict__ Wo,
    const float* __restrict__ bo, float* __restrict__ out)
{
  extern __shared__ float smem[];   // SMEM_FLOATS floats (~105 KB)

  const int t    = threadIdx.x;
  const int lane = t & 31;
  const int wave = t >> 5;        // 0..6
  const int half = lane >> 4;     // lane group 0-15 vs 16-31
  const int lr   = lane & 15;
  const int n0   = wave * 16;     // this wave's output-column tile
  const int row0 = blockIdx.x * MTILE;

  const uint32_t smem_b = (uint32_t)(uintptr_t)(void*)&smem[0];  // LDS byte base

  // Zero both W buffers once: pad cols j in [100,112) stay exactly 0 forever.
  for (int i = t; i < 2 * WBUF_FLOATS; i += NTHREADS) smem[i] = 0.0f;

  // Load x tile into h buffer 0 (cols [100,112) of h are written but never read).
  for (int i = t; i < MTILE * D; i += NTHREADS) {
    int r = i / D, c = i - r * D;
    smem[HB_BASE + r * HS + c] = x[(row0 + r) * D + c];
  }
  __syncthreads();   // zeroing must be globally done before async DMA overwrites interior

  // Stage layer 0 weights via async DMA.
  stage_async((uint64_t)(uintptr_t)W, smem_b, t);
  asm volatile("s_wait_asynccnt 0x0" ::: "memory");
  __syncthreads();

  int cur = 0;
  for (int l = 0; l < NL; ++l) {
    const int nb = cur ^ 1;
    const bool have_next = (l + 1 < NL);

    // (1) Kick off next layer's weight DMA into the alternate buffer; it runs
    //     asynchronously (ASYNCcnt) underneath the WMMA phase below.
    if (have_next) {
      stage_async((uint64_t)(uintptr_t)(W + (size_t)(l + 1) * DD),
                  smem_b + (uint32_t)nb * (WBUF_FLOATS * 4u), t);
      if (l + 2 < NL) {  // warm L2 for the layer after next (global_prefetch_b8)
        const char* pf = (const char*)(W + (size_t)(l + 2) * DD);
        __builtin_prefetch(pf + t * 128, 0, 1);
      }
    }

    // (2) This wave's 16x16 tile of h @ W^T via V_WMMA_F32_16X16X4_F32.
    //     A (16x4): lanes 0-15 hold K=k0,k0+1; lanes 16-31 hold K=k0+2,k0+3.
    //     B (4x16): B[k][n] = W[n][k] = wt[k*WS + n].
    v8f acc = {};
    const float* hc = smem + HB_BASE + cur * HBUF_FLOATS;
    const float* wc = smem + cur * WBUF_FLOATS;
    #pragma unroll 5
    for (int s = 0; s < KSTEP; ++s) {
      const int k0 = 4 * s;
      v2f a = *(const v2f*)(hc + lr * HS + k0 + 2 * half);   // aligned ds b64
      v2f bm;
      bm.x = wc[(k0 + 2 * half)     * WS + n0 + lr];
      bm.y = wc[(k0 + 2 * half + 1) * WS + n0 + lr];
      acc = __builtin_amdgcn_wmma_f32_16x16x4_f32(false, a, false, bm,
                                                  (short)0, acc, false, false);
    }

    // Bias + ReLU, write to the other h buffer. C/D layout: VGPR r -> rows r, r+8.
    float bias = (n0 + lr < D) ? b[(size_t)l * D + n0 + lr] : 0.0f;
    float* hn = smem + HB_BASE + nb * HBUF_FLOATS;
    #pragma unroll
    for (int r = 0; r < 8; ++r) {
      int   row = r + 8 * half;
      float v   = acc[r] + bias;
      hn[row * HS + n0 + lr] = v > 0.0f ? v : 0.0f;
    }

    // (3) Close this layer's DMA, then rendezvous.
    asm volatile("s_wait_asynccnt 0x0" ::: "memory");
    __syncthreads();
    cur = nb;
  }

  // Final projection [100 -> 10]: tiny, plain VALU dot products.
  if (t < MTILE * DOUT) {
    int r = t / DOUT, c = t - r * DOUT;
    const float* hrow = smem + HB_BASE + cur * HBUF_FLOATS + r * HS;
    float acc = bo[c];
    for (int k = 0; k < D; ++k) acc += hrow[k] * Wo[c * D + k];
    out[(size_t)(row0 + r) * DOUT + c] = acc;
  }
}

extern "C" void kernel_launch(void* const* d_in, const int* in_sizes, int n_in,
                              void* d_out, int out_size, void* d_ws, size_t ws_size,
                              hipStream_t stream) {
  const float* x  = (const float*)d_in[0];
  const float* W  = (const float*)d_in[1];
  const float* b  = (const float*)d_in[2];
  const float* Wo = (const float*)d_in[3];
  const float* bo = (const float*)d_in[4];
  float* out = (float*)d_out;

  const size_t shmem = (size_t)SMEM_FLOATS * sizeof(float);
  hipLaunchKernelGGL(hugenet_kernel, dim3(256 / MTILE), dim3(NTHREADS), shmem, stream,
                     x, W, b, Wo, bo, out);
}